// Model_53420803227981
// MI455X (gfx1250) — compile-verified
//
#include <hip/hip_runtime.h>

#define N_USER   150000
#define N_RECIPE 150000
#define H        64
#define UF       58
#define RF       128
#define NEDGE    2000000
#define NEL      500000

typedef float v2f __attribute__((ext_vector_type(2)));
typedef float v8f __attribute__((ext_vector_type(8)));

// D = A(16x4 f32) * B(4x16 f32) + C(16x16 f32), full-precision matrix core path.
__device__ __forceinline__ v8f wmma4(v2f a, v2f b, v8f c) {
    return __builtin_amdgcn_wmma_f32_16x16x4_f32(false, a, false, b, (short)0, c,
                                                 false, false);
}

// ---------------- utility kernels ----------------

__global__ void k_zero(float* __restrict__ p, int n) {
    int i = (blockIdx.x * blockDim.x + threadIdx.x) * 4;
    if (i < n) {
        float4 z = {0.f, 0.f, 0.f, 0.f};
        *(float4*)(p + i) = z;
    }
}

__global__ void k_degree(const int* __restrict__ us, const int* __restrict__ rd,
                         float* __restrict__ deg_u, float* __restrict__ deg_r) {
    int e = blockIdx.x * blockDim.x + threadIdx.x;
    if (e < NEDGE) {
        atomicAdd(&deg_u[us[e]], 1.0f);
        atomicAdd(&deg_r[rd[e]], 1.0f);
    }
}

__global__ void k_inv(float* __restrict__ d, int n) {
    int i = blockIdx.x * blockDim.x + threadIdx.x;
    if (i < n) d[i] = 1.0f / fmaxf(d[i], 1.0f);
}

// ---------------- edge scatter (both relations, one edge-list pass) ----------------
// 32 lanes per edge; each lane moves float2 of the 64-float row (coalesced 256B),
// accumulation via global_atomic_add_f32 which resolves in L2 (all arrays L2-resident).
__global__ void k_scatter(const int* __restrict__ us, const int* __restrict__ rd,
                          const float* __restrict__ xu, const float* __restrict__ xr,
                          float* __restrict__ sum_u, float* __restrict__ sum_r) {
    long t = (long)blockIdx.x * blockDim.x + threadIdx.x;
    int e = (int)(t >> 5);
    int part = ((int)t & 31) * 2;
    if (e < NEDGE) {
        int u = us[e];
        int r = rd[e];
        v2f mu = *(const v2f*)(xu + (long)u * H + part);  // user -> recipe message
        v2f mr = *(const v2f*)(xr + (long)r * H + part);  // recipe -> user message
        float* pr = sum_r + (long)r * H + part;
        float* pu = sum_u + (long)u * H + part;
        atomicAdd(pr,     mu.x);
        atomicAdd(pr + 1, mu.y);
        atomicAdd(pu,     mr.x);
        atomicAdd(pu + 1, mr.y);
    }
}

// ---------------- encoder: out = x @ w.T + b + emb[node_id] ----------------
// One wave = one 16x16 tile of out; 4 waves/block cover N=H=64. Grid = rows/16.
template <int K>
__global__ void k_encode(const float* __restrict__ x, const float* __restrict__ w,
                         const float* __restrict__ b, const float* __restrict__ emb,
                         const int* __restrict__ node_id, float* __restrict__ out) {
    int lane  = threadIdx.x & 31;
    int wv    = threadIdx.x >> 5;
    int mbase = blockIdx.x * 16;
    int nbase = wv * 16;
    int row   = mbase + (lane & 15);   // A-operand row for this lane
    int ncol  = nbase + (lane & 15);   // B-operand column for this lane
    int kh    = (lane >> 4) * 2;       // lane-half K offset

    v8f acc = {0.f, 0.f, 0.f, 0.f, 0.f, 0.f, 0.f, 0.f};
    for (int kb = 0; kb < K; kb += 4) {
        int k0 = kb + kh, k1 = k0 + 1;
        v2f a, bb;
        a.x  = (k0 < K) ? x[row * K + k0]  : 0.f;
        a.y  = (k1 < K) ? x[row * K + k1]  : 0.f;
        bb.x = (k0 < K) ? w[ncol * K + k0] : 0.f;   // B[k][n] = w[n][k]
        bb.y = (k1 < K) ? w[ncol * K + k1] : 0.f;
        acc = wmma4(a, bb, acc);
    }

    int half = (lane >> 4) * 8;
    int ocol = nbase + (lane & 15);
#pragma unroll
    for (int i = 0; i < 8; ++i) {
        int orow = mbase + half + i;
        int nid  = node_id[orow];
        out[orow * H + ocol] = acc[i] + b[ocol] + emb[nid * H + ocol];
    }
}

// ---------------- SAGE combine: out = (sum*inv) @ wl.T + bl + xself @ wr.T ----------------
// Fuses the mean normalization into the A-operand load and both matmuls into one
// accumulator; optional ReLU for layer 1.
__global__ void k_combine(const float* __restrict__ sum, const float* __restrict__ inv,
                          const float* __restrict__ xself,
                          const float* __restrict__ wl, const float* __restrict__ bl,
                          const float* __restrict__ wr,
                          float* __restrict__ out, int do_relu) {
    int lane  = threadIdx.x & 31;
    int wv    = threadIdx.x >> 5;
    int mbase = blockIdx.x * 16;
    int nbase = wv * 16;
    int row   = mbase + (lane & 15);
    int ncol  = nbase + (lane & 15);
    int kh    = (lane >> 4) * 2;

    float s = inv[row];  // 1 / max(count, 1)
    v8f acc = {0.f, 0.f, 0.f, 0.f, 0.f, 0.f, 0.f, 0.f};
#pragma unroll
    for (int kb = 0; kb < H; kb += 4) {
        int k0 = kb + kh;
        v2f a1 = *(const v2f*)(sum + row * H + k0);
        a1.x *= s; a1.y *= s;
        v2f b1 = *(const v2f*)(wl + ncol * H + k0);
        acc = wmma4(a1, b1, acc);

        v2f a2 = *(const v2f*)(xself + row * H + k0);
        v2f b2 = *(const v2f*)(wr + ncol * H + k0);
        acc = wmma4(a2, b2, acc);
    }

    int half = (lane >> 4) * 8;
    int ocol = nbase + (lane & 15);
#pragma unroll
    for (int i = 0; i < 8; ++i) {
        int orow = mbase + half + i;
        float v = acc[i] + bl[ocol];
        if (do_relu) v = fmaxf(v, 0.f);
        out[orow * H + ocol] = v;
    }
}

// ---------------- edge dot classifier ----------------
__global__ void k_edgedot(const int* __restrict__ eli, const float* __restrict__ hu,
                          const float* __restrict__ hr, float* __restrict__ out) {
    int e = blockIdx.x * blockDim.x + threadIdx.x;
    if (e < NEL) {
        int i = eli[e];
        int j = eli[NEL + e];
        const float4* a = (const float4*)(hu + (long)i * H);
        const float4* b = (const float4*)(hr + (long)j * H);
        float acc = 0.f;
#pragma unroll
        for (int t = 0; t < 16; ++t) {
            float4 av = a[t], bv = b[t];
            acc += av.x * bv.x + av.y * bv.y + av.z * bv.z + av.w * bv.w;
        }
        out[e] = acc;
    }
}

// ---------------- driver ----------------
extern "C" void kernel_launch(void* const* d_in, const int* in_sizes, int n_in,
                              void* d_out, int out_size, void* d_ws, size_t ws_size,
                              hipStream_t stream) {
    const float* x_user   = (const float*)d_in[0];
    const float* x_recipe = (const float*)d_in[1];
    const int*   user_id  = (const int*)d_in[2];
    const int*   rec_id   = (const int*)d_in[3];
    const int*   eidx     = (const int*)d_in[4];
    const int*   eli      = (const int*)d_in[5];
    const float* u_w      = (const float*)d_in[6];
    const float* u_b      = (const float*)d_in[7];
    const float* r_w      = (const float*)d_in[8];
    const float* r_b      = (const float*)d_in[9];
    const float* u_emb    = (const float*)d_in[10];
    const float* r_emb    = (const float*)d_in[11];
    const float* c1ra_wl  = (const float*)d_in[12];
    const float* c1ra_bl  = (const float*)d_in[13];
    const float* c1ra_wr  = (const float*)d_in[14];
    const float* c1rv_wl  = (const float*)d_in[15];
    const float* c1rv_bl  = (const float*)d_in[16];
    const float* c1rv_wr  = (const float*)d_in[17];
    const float* c2ra_wl  = (const float*)d_in[18];
    const float* c2ra_bl  = (const float*)d_in[19];
    const float* c2ra_wr  = (const float*)d_in[20];
    const float* c2rv_wl  = (const float*)d_in[21];
    const float* c2rv_bl  = (const float*)d_in[22];
    const float* c2rv_wr  = (const float*)d_in[23];

    float* ws = (float*)d_ws;
    const size_t NF = (size_t)N_USER * H;  // 9.6M floats per node-feature buffer
    float* xu    = ws;            // hu2 overwrites this in layer 2
    float* xr    = ws + 1 * NF;   // hr2 overwrites this in layer 2
    float* sum_u = ws + 2 * NF;
    float* sum_r = ws + 3 * NF;
    float* hu    = ws + 4 * NF;
    float* hr    = ws + 5 * NF;
    float* inv_u = ws + 6 * NF;          // 150016 floats
    float* inv_r = inv_u + 150016;       // 150016 floats

    const int* us = eidx;          // edge_index[0]: user source
    const int* rd = eidx + NEDGE;  // edge_index[1]: recipe dest

    dim3 bEnc(128), gEnc(N_USER / 16);   // 9375 blocks, exact -> EXEC all-ones

    // Reciprocal degrees (shared by both layers).
    const int ndeg = 150016 * 2;
    k_zero<<<(ndeg / 4 + 255) / 256, 256, 0, stream>>>(inv_u, ndeg);
    k_degree<<<(NEDGE + 255) / 256, 256, 0, stream>>>(us, rd, inv_u, inv_r);
    k_inv<<<(ndeg + 255) / 256, 256, 0, stream>>>(inv_u, ndeg);

    // Input encoders (WMMA).
    k_encode<UF><<<gEnc, bEnc, 0, stream>>>(x_user, u_w, u_b, u_emb, user_id, xu);
    k_encode<RF><<<gEnc, bEnc, 0, stream>>>(x_recipe, r_w, r_b, r_emb, rec_id, xr);

    const int nsum = (int)(2 * NF);
    const int scatBlocks = (int)(((long)NEDGE * 32) / 256);  // 250000, exact

    // Layer 1: scatter + combine (+ReLU).
    k_zero<<<(nsum / 4 + 255) / 256, 256, 0, stream>>>(sum_u, nsum);
    k_scatter<<<scatBlocks, 256, 0, stream>>>(us, rd, xu, xr, sum_u, sum_r);
    k_combine<<<gEnc, bEnc, 0, stream>>>(sum_r, inv_r, xr, c1ra_wl, c1ra_bl, c1ra_wr, hr, 1);
    k_combine<<<gEnc, bEnc, 0, stream>>>(sum_u, inv_u, xu, c1rv_wl, c1rv_bl, c1rv_wr, hu, 1);

    // Layer 2: scatter + combine (no ReLU), outputs reuse xu/xr buffers.
    k_zero<<<(nsum / 4 + 255) / 256, 256, 0, stream>>>(sum_u, nsum);
    k_scatter<<<scatBlocks, 256, 0, stream>>>(us, rd, hu, hr, sum_u, sum_r);
    k_combine<<<gEnc, bEnc, 0, stream>>>(sum_r, inv_r, hr, c2ra_wl, c2ra_bl, c2ra_wr, xr, 0);
    k_combine<<<gEnc, bEnc, 0, stream>>>(sum_u, inv_u, hu, c2rv_wl, c2rv_bl, c2rv_wr, xu, 0);

    // Edge dot-product classifier.
    k_edgedot<<<(NEL + 255) / 256, 256, 0, stream>>>(eli, xu, xr, (float*)d_out);
}